// CondGIN_81140522156092
// MI455X (gfx1250) — compile-verified
//
#include <hip/hip_runtime.h>

#define DD 96
#define CHN 5
#define CDN 7
#define LATN 64
#define EPS 1e-5f

typedef __attribute__((ext_vector_type(2))) float v2f;
typedef __attribute__((ext_vector_type(8))) float v8f;

// ---------------- utility kernels ----------------

__global__ void copy4_kernel(const float4* __restrict__ src, float4* __restrict__ dst, int n4) {
  int i = blockIdx.x * blockDim.x + threadIdx.x;
  if (i < n4) dst[i] = src[i];
}

__global__ void zero_kernel(float* __restrict__ p, int n) {
  int i = blockIdx.x * blockDim.x + threadIdx.x;
  if (i < n) p[i] = 0.0f;
}

// scatter-add h[src[e]] into agg[dst[e]] ; one thread = one edge x 4-float chunk
__global__ void edge_scatter_kernel(const int* __restrict__ srcIdx, const int* __restrict__ dstIdx,
                                    const float* __restrict__ h, float* __restrict__ agg, int E) {
  int idx = blockIdx.x * blockDim.x + threadIdx.x;
  if (idx >= E * 24) return;
  int e = idx / 24;
  int c = (idx % 24) * 4;
  int s = srcIdx[e];
  int d = dstIdx[e];
  const float4 v = *(const float4*)(h + (size_t)s * DD + c);
  float* p = agg + (size_t)d * DD + c;
  atomicAdd(p + 0, v.x);
  atomicAdd(p + 1, v.y);
  atomicAdd(p + 2, v.z);
  atomicAdd(p + 3, v.w);
}

// pooled[batch[i]] += h[i] ; one thread = one node x 4-float chunk
__global__ void pool_scatter_kernel(const int* __restrict__ batch, const float* __restrict__ h,
                                    float* __restrict__ pooled, int N) {
  int idx = blockIdx.x * blockDim.x + threadIdx.x;
  if (idx >= N * 24) return;
  int i = idx / 24;
  int c = (idx % 24) * 4;
  int g = batch[i];
  const float4 v = *(const float4*)(h + (size_t)i * DD + c);
  float* p = pooled + (size_t)g * DD + c;
  atomicAdd(p + 0, v.x);
  atomicAdd(p + 1, v.y);
  atomicAdd(p + 2, v.z);
  atomicAdd(p + 3, v.w);
}

// ---------------- WMMA GEMM: out = post( in[N x 96] @ W[96 x 96] + bias ) ----------------
// post = leaky_relu(0.2) then optional BN-affine (eval mode) when apply_bn != 0.
// One wave computes a 16-row x 96-col tile with V_WMMA_F32_16X16X4_F32 (exact f32 math).
// W is staged in LDS pre-swizzled into per-lane WMMA B-fragment order so each
// fragment is one aligned ds_load_b64 into an even VGPR pair (no re-pairing movs).
__global__ __launch_bounds__(256)
void gemm96_kernel(const float* __restrict__ in, const float* __restrict__ W,
                   const float* __restrict__ bias,
                   const float* __restrict__ ggain, const float* __restrict__ gbeta,
                   const float* __restrict__ gmean, const float* __restrict__ gvar,
                   float* __restrict__ out, int n, int apply_bn) {
  // sB[((kb/4)*6 + t)*64 + lane*2 + v] = W[(kb + koff(lane) + v)*96 + t*16 + (lane&15)]
  __shared__ float sB[24 * 6 * 64];  // 36 KB of the 320 KB/WGP LDS
  for (int i = threadIdx.x; i < 24 * 6 * 32; i += blockDim.x) {
    const int l2 = i & 31;          // lane this entry feeds
    const int rest = i >> 5;        // kblk*6 + t
    const int t = rest % 6;
    const int kblk = rest / 6;
    const int k0 = kblk * 4 + ((l2 & 16) ? 2 : 0);
    const int col = t * 16 + (l2 & 15);
    sB[i * 2 + 0] = W[k0 * DD + col];
    sB[i * 2 + 1] = W[(k0 + 1) * DD + col];
  }
  __syncthreads();

  const int lane = threadIdx.x & 31;
  const int wave = threadIdx.x >> 5;
  const int rowBase = (blockIdx.x * 8 + wave) * 16;
  if (rowBase >= n) return;  // wave-uniform exit: EXEC stays all-ones in live waves

  const int mrow = rowBase + (lane & 15);
  const int m = (mrow < n) ? mrow : (n - 1);  // clamp loads; stores predicated below
  const int koff = (lane < 16) ? 0 : 2;       // A lane split on K (16x4 f32 layout)
  const int nlo = lane & 15;

  v8f acc[6];
  #pragma unroll
  for (int t = 0; t < 6; ++t) acc[t] = {};

  const float* arow = in + (size_t)m * DD;
  const float* blane = sB + lane * 2;
  for (int kb = 0; kb < DD; kb += 4) {
    v2f a;
    a.x = arow[kb + koff];
    a.y = arow[kb + koff + 1];
    const float* bk = blane + (kb >> 2) * (6 * 64);
    #pragma unroll
    for (int t = 0; t < 6; ++t) {
      const v2f b = *(const v2f*)(bk + t * 64);  // single aligned ds_load_b64
      // (neg_a, A, neg_b, B, c_mod, C, reuse_a, reuse_b)
      acc[t] = __builtin_amdgcn_wmma_f32_16x16x4_f32(
          false, a, false, b, (short)0, acc[t], false, false);
    }
  }

  const int rowHi = (lane & 16) ? 8 : 0;  // D layout: lanes 16-31 hold M = r+8
  #pragma unroll
  for (int t = 0; t < 6; ++t) {
    const int col = t * 16 + nlo;  // one output column per (lane, tile)
    const float bb = bias[col];
    float s = 1.0f, mu = 0.0f, bt = 0.0f;
    if (apply_bn) {
      s = ggain[col] * rsqrtf(gvar[col] + EPS);
      mu = gmean[col];
      bt = gbeta[col];
    }
    #pragma unroll
    for (int r = 0; r < 8; ++r) {
      const int row = rowBase + rowHi + r;
      if (row < n) {
        float v = acc[t][r] + bb;
        v = (v > 0.0f) ? v : 0.2f * v;  // leaky_relu 0.2
        if (apply_bn) v = (v - mu) * s + bt;
        out[(size_t)row * DD + col] = v;
      }
    }
  }
}

// ---------------- tail: cond MLP + concat + BN + FC (tiny; scalar) ----------------
__global__ void tail_kernel(const float* __restrict__ cond, const float* __restrict__ pooled,
                            const float* __restrict__ cW1, const float* __restrict__ cb1,
                            const float* __restrict__ cg, const float* __restrict__ cbeta,
                            const float* __restrict__ cmean, const float* __restrict__ cvar,
                            const float* __restrict__ cW2, const float* __restrict__ cb2,
                            const float* __restrict__ bn_g, const float* __restrict__ bn_b,
                            const float* __restrict__ bn_mean, const float* __restrict__ bn_var,
                            const float* __restrict__ fc_W, const float* __restrict__ fc_b,
                            float* __restrict__ out, int G) {
  int idx = blockIdx.x * blockDim.x + threadIdx.x;
  if (idx >= G * LATN) return;
  int g = idx / LATN;
  int lat = idx % LATN;

  // c1 = relu(BN(cond @ cW1 + cb1))
  float c1[CHN];
  #pragma unroll
  for (int j = 0; j < CHN; ++j) {
    float t = cb1[j];
    #pragma unroll
    for (int k = 0; k < CDN; ++k) t += cond[g * CDN + k] * cW1[k * CHN + j];
    t = (t - cmean[j]) * (cg[j] * rsqrtf(cvar[j] + EPS)) + cbeta[j];
    c1[j] = t > 0.0f ? t : 0.0f;
  }
  // c2 = relu(c1 @ cW2 + cb2)
  float c2[CHN];
  #pragma unroll
  for (int j = 0; j < CHN; ++j) {
    float t = cb2[j];
    #pragma unroll
    for (int k = 0; k < CHN; ++k) t += c1[k] * cW2[k * CHN + j];
    c2[j] = t > 0.0f ? t : 0.0f;
  }
  // out = BN(concat(c2, pooled)) @ fc_W + fc_b
  float o = fc_b[lat];
  for (int j = 0; j < CHN + DD; ++j) {
    float v = (j < CHN) ? c2[j] : pooled[(size_t)g * DD + (j - CHN)];
    v = (v - bn_mean[j]) * (bn_g[j] * rsqrtf(bn_var[j] + EPS)) + bn_b[j];
    o += v * fc_W[j * LATN + lat];
  }
  out[(size_t)g * LATN + lat] = o;
}

// ---------------- launcher ----------------
extern "C" void kernel_launch(void* const* d_in, const int* in_sizes, int n_in,
                              void* d_out, int out_size, void* d_ws, size_t ws_size,
                              hipStream_t stream) {
  const float* x        = (const float*)d_in[0];
  const float* cond     = (const float*)d_in[1];
  const int*   eidx     = (const int*)d_in[2];
  const int*   batch    = (const int*)d_in[3];
  const float* conv_W1  = (const float*)d_in[4];
  const float* conv_b1  = (const float*)d_in[5];
  const float* conv_g   = (const float*)d_in[6];
  const float* conv_be  = (const float*)d_in[7];
  const float* conv_mu  = (const float*)d_in[8];
  const float* conv_var = (const float*)d_in[9];
  const float* conv_W2  = (const float*)d_in[10];
  const float* conv_b2  = (const float*)d_in[11];
  const float* cW1      = (const float*)d_in[12];
  const float* cb1      = (const float*)d_in[13];
  const float* cg       = (const float*)d_in[14];
  const float* cbeta    = (const float*)d_in[15];
  const float* cmean    = (const float*)d_in[16];
  const float* cvar     = (const float*)d_in[17];
  const float* cW2      = (const float*)d_in[18];
  const float* cb2      = (const float*)d_in[19];
  const float* bn_g     = (const float*)d_in[20];
  const float* bn_b     = (const float*)d_in[21];
  const float* bn_mean  = (const float*)d_in[22];
  const float* bn_var   = (const float*)d_in[23];
  const float* fc_W     = (const float*)d_in[24];
  const float* fc_b     = (const float*)d_in[25];

  const int N = in_sizes[0] / DD;
  const int G = in_sizes[1] / CDN;
  const int E = in_sizes[2] / 2;
  const int L = in_sizes[4] / (DD * DD);
  const int* srcIdx = eidx;
  const int* dstIdx = eidx + E;

  const size_t ND = (size_t)N * DD;
  float* h      = (float*)d_ws;
  float* agg    = h + ND;
  float* tmp    = agg + ND;
  float* pooled = tmp + ND;

  const int n4 = (int)(ND / 4);
  const int eThreads = E * 24;
  const int gemmGrid = (N + 127) / 128;  // 8 waves/block, 16 rows/wave

  for (int l = 0; l < L; ++l) {
    const float* hin = (l == 0) ? x : h;
    // agg = hin  (self term of GIN eps=0), then += neighbor sum
    copy4_kernel<<<(n4 + 255) / 256, 256, 0, stream>>>((const float4*)hin, (float4*)agg, n4);
    edge_scatter_kernel<<<(eThreads + 255) / 256, 256, 0, stream>>>(srcIdx, dstIdx, hin, agg, E);
    // tmp = BN(leaky(agg @ W1 + b1))
    gemm96_kernel<<<gemmGrid, 256, 0, stream>>>(
        agg, conv_W1 + (size_t)l * DD * DD, conv_b1 + (size_t)l * DD,
        conv_g + (size_t)l * DD, conv_be + (size_t)l * DD,
        conv_mu + (size_t)l * DD, conv_var + (size_t)l * DD, tmp, N, 1);
    // h = leaky(tmp @ W2 + b2)
    gemm96_kernel<<<gemmGrid, 256, 0, stream>>>(
        tmp, conv_W2 + (size_t)l * DD * DD, conv_b2 + (size_t)l * DD,
        nullptr, nullptr, nullptr, nullptr, h, N, 0);
  }

  const int gd = G * DD;
  zero_kernel<<<(gd + 255) / 256, 256, 0, stream>>>(pooled, gd);
  const int pThreads = N * 24;
  pool_scatter_kernel<<<(pThreads + 255) / 256, 256, 0, stream>>>(batch, h, pooled, N);

  tail_kernel<<<(G * LATN + 255) / 256, 256, 0, stream>>>(
      cond, pooled, cW1, cb1, cg, cbeta, cmean, cvar, cW2, cb2,
      bn_g, bn_b, bn_mean, bn_var, fc_W, fc_b, (float*)d_out, G);
}